// SNN_Net_28071906246997
// MI455X (gfx1250) — compile-verified
//
#include <hip/hip_runtime.h>

typedef __attribute__((ext_vector_type(2))) float v2f;
typedef __attribute__((ext_vector_type(8))) float v8f;

#define L_TAU   0.8f
#define F_L2E   1.44269504088896340736f   // log2(e)
#define F_LN2   0.69314718055994530942f   // ln(2)

// ---------------------------------------------------------------------------
// Kernel 1: SNU recurrence. One wave32 handles 16 batch elements.
//   A (16x4 f32): row m = batch element b0+m, K = the 4 inputs at time t.
//       Lane layout: lanes 0-15 -> VGPR0=K0, VGPR1=K1 ; lanes 16-31 -> K2,K3.
//   B (4x16 f32): B[k][n] = W[n][k] (n<3), 0 otherwise. Constant across t.
//   D (16x16 f32): z tile; state s / (1-o) / acc kept in the same layout so
//       the whole recurrence is elementwise v8f math — no cross-lane traffic.
// ---------------------------------------------------------------------------
__global__ __launch_bounds__(256) void snn_forward_wmma(
    const float* __restrict__ x,      // [T, B, 4]
    const float* __restrict__ W,      // [3, 4]
    const float* __restrict__ bias,   // [3]
    float* __restrict__ sum_out,      // [B, 3]
    int B, int T)
{
    const int lane = threadIdx.x & 31;
    const int wave = blockIdx.x * 8 + (threadIdx.x >> 5);
    const int b0   = wave * 16;
    const int m    = lane & 15;   // A: row index | B/D: column index
    const int half = lane >> 4;   // K-half selector
    const int k0   = half * 2;

    // B matrix operand: B[k][n] = W[n][k]; zero for unused columns n>=3.
    float w0 = 0.0f, w1 = 0.0f, bb = 0.0f;
    if (m < 3) {
        w0 = W[m * 4 + k0];
        w1 = W[m * 4 + k0 + 1];
        bb = bias[m];             // bias for this lane's column, all 8 rows
    }
    v2f bmat; bmat.x = w0; bmat.y = w1;

    v8f s   = {0,0,0,0,0,0,0,0};  // membrane state
    v8f dv  = {1,1,1,1,1,1,1,1};  // 1 - o_prev
    v8f acc = {0,0,0,0,0,0,0,0};  // sum of spikes

    const size_t strideT = (size_t)B * 4;                    // floats per step
    const float* xp = x + (size_t)(b0 + m) * 4 + k0;         // this lane's slot
    v2f a = *(const v2f*)xp;                                 // t = 0 (8B aligned)

    for (int t = 0; t < T; ++t) {
        const int tn = (t + 1 < T) ? (t + 1) : t;            // double buffer
        v2f an = *(const v2f*)(xp + (size_t)tn * strideT);

        v8f c = {0,0,0,0,0,0,0,0};
        // z = x_t @ W^T   (16x16 tile, columns 0..2 meaningful)
        v8f z = __builtin_amdgcn_wmma_f32_16x16x4_f32(
            /*neg_a=*/false, a, /*neg_b=*/false, bmat,
            /*c_mod=*/(short)0, c, /*reuse_a=*/false, /*reuse_b=*/false);

#pragma unroll
        for (int i = 0; i < 8; ++i) {
            float si = z[i] + L_TAU * s[i] * dv[i];          // leak term
            si = fmaxf(si, 0.0f);                            // relu
            float e  = __builtin_amdgcn_exp2f(-F_L2E * (si + bb));
            float oi = __builtin_amdgcn_rcpf(1.0f + e);      // sigmoid
            s[i]   = si;
            dv[i]  = 1.0f - oi;
            acc[i] += oi;
        }
        a = an;
    }

    // Scatter accumulator: lane (column j<3, half h) holds rows 8h..8h+7.
    if (m < 3) {
#pragma unroll
        for (int i = 0; i < 8; ++i)
            sum_out[(size_t)(b0 + half * 8 + i) * 3 + m] = acc[i];
    }
}

// ---------------------------------------------------------------------------
// Kernel 2: per-element cross-entropy + accuracy, block-level partial sums.
// ---------------------------------------------------------------------------
__global__ __launch_bounds__(256) void snn_loss_partial(
    const float* __restrict__ sum_out, const int* __restrict__ y,
    float* __restrict__ part)
{
    __shared__ float sl[256];
    __shared__ float sa[256];
    const int tid = threadIdx.x;
    const int b   = blockIdx.x * 256 + tid;

    const float s0 = sum_out[(size_t)b * 3 + 0];
    const float s1 = sum_out[(size_t)b * 3 + 1];
    const float s2 = sum_out[(size_t)b * 3 + 2];

    const float mx = fmaxf(s0, fmaxf(s1, s2));
    const float se = __builtin_amdgcn_exp2f((s0 - mx) * F_L2E)
                   + __builtin_amdgcn_exp2f((s1 - mx) * F_L2E)
                   + __builtin_amdgcn_exp2f((s2 - mx) * F_L2E);
    const float lse = mx + __builtin_amdgcn_logf(se) * F_LN2;

    const int yi = y[b];
    const float sy = (yi == 0) ? s0 : ((yi == 1) ? s1 : s2);

    int pred = 0; float best = s0;
    if (s1 > best) { pred = 1; best = s1; }
    if (s2 > best) { pred = 2; }

    sl[tid] = lse - sy;
    sa[tid] = (pred == yi) ? 1.0f : 0.0f;
    __syncthreads();

    for (int off = 128; off > 0; off >>= 1) {
        if (tid < off) { sl[tid] += sl[tid + off]; sa[tid] += sa[tid + off]; }
        __syncthreads();
    }
    if (tid == 0) {
        part[(size_t)blockIdx.x * 2 + 0] = sl[0];
        part[(size_t)blockIdx.x * 2 + 1] = sa[0];
    }
}

// ---------------------------------------------------------------------------
// Kernel 3: deterministic final reduction + mean.
// ---------------------------------------------------------------------------
__global__ __launch_bounds__(256) void snn_finalize(
    const float* __restrict__ part, int nPart, float* __restrict__ out, int B)
{
    __shared__ float sl[256];
    __shared__ float sa[256];
    const int tid = threadIdx.x;
    float l = 0.0f, a = 0.0f;
    for (int i = tid; i < nPart; i += 256) {
        l += part[(size_t)i * 2 + 0];
        a += part[(size_t)i * 2 + 1];
    }
    sl[tid] = l; sa[tid] = a;
    __syncthreads();
    for (int off = 128; off > 0; off >>= 1) {
        if (tid < off) { sl[tid] += sl[tid + off]; sa[tid] += sa[tid + off]; }
        __syncthreads();
    }
    if (tid == 0) {
        const float inv = 1.0f / (float)B;
        out[0] = sl[0] * inv;   // loss
        out[1] = sa[0] * inv;   // accuracy
    }
}

// ---------------------------------------------------------------------------
extern "C" void kernel_launch(void* const* d_in, const int* in_sizes, int n_in,
                              void* d_out, int out_size, void* d_ws, size_t ws_size,
                              hipStream_t stream)
{
    (void)n_in; (void)out_size; (void)ws_size;
    const float* x    = (const float*)d_in[0];   // [T, B, 4]
    const int*   y    = (const int*)  d_in[1];   // [B]
    const float* W    = (const float*)d_in[2];   // [3, 4]
    const float* bias = (const float*)d_in[3];   // [3]
    float* out = (float*)d_out;                  // [B*3] ++ [loss, acc]

    const int B = in_sizes[1];
    const int T = in_sizes[0] / (B * 4);

    const int fwdBlocks = B / 128;               // 8 waves/block, 16 elems/wave
    snn_forward_wmma<<<fwdBlocks, 256, 0, stream>>>(x, W, bias, out, B, T);

    const int redBlocks = B / 256;
    float* part = (float*)d_ws;                  // redBlocks * 2 floats
    snn_loss_partial<<<redBlocks, 256, 0, stream>>>(out, y, part);

    snn_finalize<<<1, 256, 0, stream>>>(part, redBlocks, out + (size_t)B * 3, B);
}